// LightGCNLayer_65137474011642
// MI455X (gfx1250) — compile-verified
//
#include <hip/hip_runtime.h>

// LightGCN layer: out[dst] += w * x[src], D = 128 floats per row.
// Wave-per-edge; double-buffered async-LDS staging of edge metadata
// (ASYNCcnt partial waits); L2-resident non-returning f32 atomics.

#define D_FEAT 128
#define BLOCK  256                  // 8 wave32 per block
#define WAVES  (BLOCK / 32)
#define TILE   1024                 // edges per staged tile
#define TARGET_BLOCKS 768

// ---- gfx1250 async global->LDS copy (ASYNCcnt path), with safe fallback ----
#if defined(__has_builtin)
#  if __has_builtin(__builtin_amdgcn_global_load_async_to_lds_b128)
#    define HAS_ASYNC_LDS 1
#  endif
#endif
#ifndef HAS_ASYNC_LDS
#  define HAS_ASYNC_LDS 0
#endif

#if HAS_ASYNC_LDS
typedef __attribute__((ext_vector_type(4))) int v4i;
typedef __attribute__((address_space(1))) v4i g_v4i;
typedef __attribute__((address_space(3))) v4i l_v4i;

__device__ __forceinline__ void async_copy16(const void* g, void* l) {
  // per-lane 16B global -> LDS, one wave instruction = 512B, tracked by ASYNCcnt
  __builtin_amdgcn_global_load_async_to_lds_b128((g_v4i*)g, (l_v4i*)l, 0, 0);
}
#if __has_builtin(__builtin_amdgcn_s_wait_asynccnt)
__device__ __forceinline__ void async_wait_le0() { __builtin_amdgcn_s_wait_asynccnt(0); }
__device__ __forceinline__ void async_wait_le5() { __builtin_amdgcn_s_wait_asynccnt(5); }
#else
__device__ __forceinline__ void async_wait_le0() { asm volatile("s_wait_asynccnt 0" ::: "memory"); }
__device__ __forceinline__ void async_wait_le5() { asm volatile("s_wait_asynccnt 5" ::: "memory"); }
#endif
#else
__device__ __forceinline__ void async_copy16(const void* g, void* l) {
  *(float4*)l = *(const float4*)g;            // synchronous fallback
}
__device__ __forceinline__ void async_wait_le0() {}
__device__ __forceinline__ void async_wait_le5() {}
#endif

// ---------------------------------------------------------------------------

__global__ void __launch_bounds__(BLOCK) zero_out_kernel(float4* __restrict__ o, int n4) {
  int i = blockIdx.x * blockDim.x + threadIdx.x;
  if (i < n4) o[i] = make_float4(0.f, 0.f, 0.f, 0.f);
}

__global__ void __launch_bounds__(BLOCK) lightgcn_scatter_kernel(
    const float* __restrict__ x,
    const float* __restrict__ ew,
    const long long* __restrict__ esrc,
    const long long* __restrict__ edst,
    float* __restrict__ out,
    int n_edges, int tiles_per_block)
{
  // Double-buffered edge-metadata tiles: 2 * (8 + 8 + 4) KB = 40 KB LDS.
  __shared__ __align__(16) long long s_src[2][TILE];
  __shared__ __align__(16) long long s_dst[2][TILE];
  __shared__ __align__(16) float     s_w[2][TILE];

  const int lane = threadIdx.x & 31;
  const int wv   = threadIdx.x >> 5;
  const int n_tiles = (n_edges + TILE - 1) / TILE;

  const int t0    = blockIdx.x * tiles_per_block;
  const int t_end = min(t0 + tiles_per_block, n_tiles);
  if (t0 >= n_tiles) return;

  // Issue one tile's staging: 5 async wave-instructions (2 + 2 + 1).
  auto stage_async = [&](int buf, int base) {
    for (int i = threadIdx.x; i < TILE / 2; i += BLOCK) {   // 2 iters/thread
      async_copy16(esrc + base + i * 2, &s_src[buf][i * 2]);
      async_copy16(edst + base + i * 2, &s_dst[buf][i * 2]);
    }
    for (int i = threadIdx.x; i < TILE / 4; i += BLOCK) {   // 1 iter/thread
      async_copy16(ew + base + i * 4, &s_w[buf][i * 4]);
    }
  };

  int buf = 0;

  // Prime the pipeline with the first tile (if full).
  if (n_edges - t0 * TILE >= TILE) stage_async(buf, t0 * TILE);

  for (int t = t0; t < t_end; ++t) {
    const int base  = t * TILE;
    const int count = min(TILE, n_edges - base);
    const bool cur_full = (count == TILE);

    // Kick off the next tile into the other buffer while we consume this one.
    bool next_full = false;
    if (t + 1 < t_end) {
      const int nbase = (t + 1) * TILE;
      next_full = (n_edges - nbase >= TILE);
      if (next_full) stage_async(buf ^ 1, nbase);
    }

    if (cur_full) {
      // Async ops complete in order: <=5 outstanding means current tile landed.
      if (next_full) async_wait_le5(); else async_wait_le0();
    } else {
      // Only the single global-last partial tile takes this path; no asyncs
      // can be outstanding when we get here.
      for (int i = threadIdx.x; i < count; i += BLOCK) {
        s_src[buf][i] = esrc[base + i];
        s_dst[buf][i] = edst[base + i];
        s_w[buf][i]   = ew[base + i];
      }
    }
    __syncthreads();

    // One wave per edge: 32 lanes x float4 = full 128-float (512B) row.
#pragma unroll 2
    for (int e = wv; e < count; e += WAVES) {
      const long long s = s_src[buf][e];
      const long long d = s_dst[buf][e];
      const float     w = s_w[buf][e];

      const int en = e + WAVES;
      if (en < count) {
        __builtin_prefetch(x + (size_t)s_src[buf][en] * D_FEAT + lane * 4, 0, 0);
      }

      const float4 v = ((const float4*)(x + (size_t)s * D_FEAT))[lane];
      float* o = out + (size_t)d * D_FEAT + lane * 4;
      atomicAdd(o + 0, v.x * w);
      atomicAdd(o + 1, v.y * w);
      atomicAdd(o + 2, v.z * w);
      atomicAdd(o + 3, v.w * w);
    }
    __syncthreads();   // all reads of this buffer done before it is restaged
    buf ^= 1;
  }
}

extern "C" void kernel_launch(void* const* d_in, const int* in_sizes, int n_in,
                              void* d_out, int out_size, void* d_ws, size_t ws_size,
                              hipStream_t stream) {
  const float*     x    = (const float*)d_in[0];       // (N, 128) f32
  const float*     ew   = (const float*)d_in[1];       // (E,)     f32
  const long long* eidx = (const long long*)d_in[2];   // (2, E)   i64

  const int n_edges = in_sizes[1];
  const long long* esrc = eidx;             // row 0
  const long long* edst = eidx + n_edges;   // row 1
  float* out = (float*)d_out;

  // Zero the accumulator (harness poisons d_out; atomics need clean zeros).
  const int n4 = out_size / 4;
  zero_out_kernel<<<(n4 + BLOCK - 1) / BLOCK, BLOCK, 0, stream>>>((float4*)out, n4);

  const int n_tiles = (n_edges + TILE - 1) / TILE;
  int tpb = (n_tiles + TARGET_BLOCKS - 1) / TARGET_BLOCKS;
  if (tpb < 1) tpb = 1;
  const int n_blocks = (n_tiles + tpb - 1) / tpb;
  lightgcn_scatter_kernel<<<n_blocks, BLOCK, 0, stream>>>(x, ew, esrc, edst, out,
                                                          n_edges, tpb);
}